// MaceGNN_16999480557842
// MI455X (gfx1250) — compile-verified
//
#include <hip/hip_runtime.h>
#include <hip/hip_bf16.h>

// ---------------------------------------------------------------------------
// MACE GNN forward for MI455X (gfx1250, wave32).
//  * All GEMMs via V_WMMA_F32_16X16X4_F32 (fp32 end-to-end; K dims all %4==0,
//    N dims all %64==0 -> one wave computes a 16x64 strip = 4 WMMA tiles,
//    reusing one A fragment (8B vector load) across 4 B tiles).
//  * global_prefetch on the streaming B operand (W1 = 165MB is the bandwidth
//    hot spot: ~7us at 23.3TB/s; everything else fits the 192MB L2).
//  * Deterministic segment-sum: chunked counting sort of edges by receiver,
//    then per-node ordered accumulation (graph replay => bitwise stable).
// ---------------------------------------------------------------------------

#define NGR   16
#define AA    400
#define NN    6400          // NGR*AA
#define EE    65536
#define CC    64
#define D1    79800         // A*(A-1)/2
#define DH    640
#define DIN   80440         // D1+DH
#define NCH   64            // chunks for counting sort
#define CHSZ  1024          // EE/NCH

typedef float v2f __attribute__((ext_vector_type(2)));
typedef float v8f __attribute__((ext_vector_type(8)));

// ------------------------------ spherical harmonics ------------------------
__global__ void k_sph(const float* __restrict__ ev, float* __restrict__ Y) {
  int e = blockIdx.x * blockDim.x + threadIdx.x;
  float vx = ev[e], vy = ev[EE + e], vz = ev[2 * EE + e];
  float nrm = sqrtf(vx * vx + vy * vy + vz * vz) + 1e-9f;
  float x = vx / nrm, y = vy / nrm, z = vz / nrm;
  const float S3 = 1.7320508075688772f;
  const float C95 = 0.7905694150420949f;   // sqrt(5/8)
  const float S15 = 3.872983346207417f;    // sqrt(15)
  const float C38 = 0.6123724356957945f;   // sqrt(3/8)
  const float H15 = 1.9364916731037085f;   // 0.5*sqrt(15)
  float* o = Y + (size_t)e * 16;
  o[0] = 1.0f; o[1] = x; o[2] = y; o[3] = z;
  o[4] = S3 * x * y;
  o[5] = S3 * y * z;
  o[6] = 0.5f * (3.0f * z * z - 1.0f);
  o[7] = S3 * x * z;
  o[8] = 0.5f * S3 * (x * x - y * y);
  o[9] = C95 * y * (3.0f * x * x - y * y);
  o[10] = S15 * x * y * z;
  o[11] = C38 * y * (5.0f * z * z - 1.0f);
  o[12] = 0.5f * z * (5.0f * z * z - 3.0f);
  o[13] = C38 * x * (5.0f * z * z - 1.0f);
  o[14] = H15 * z * (x * x - y * y);
  o[15] = C95 * x * (x * x - 3.0f * y * y);
}

// ------------------------------ generic WMMA f32 GEMM ----------------------
// One wave (32 threads) computes a 16x64 strip (4 WMMA 16x16 tiles) of
// C = act(A@B [+ A2@B2] + bias).
// A: row-major [r][k] at A[r*lda + k] (unit k-stride -> 8B vector load).
// A2: row-major with element k-stride as2 (for the interleaved vector skips).
// B/B2: row-major [k][n].  C element [r][n] at C[r*ldc + coff + n*cstride].
// ISA layout (7.12.2): A f32 16x4 -> lanes0-15 K={0,1}, lanes16-31 K={2,3};
// C/D: vgpr i -> row i+8*half, col lane&15.
__global__ void k_gemm4(const float* __restrict__ A, int lda,
                        const float* __restrict__ B, int ldb,
                        const float* __restrict__ A2, int lda2, int as2,
                        const float* __restrict__ B2,
                        const float* __restrict__ bias,
                        float* __restrict__ C, int ldc, int coff, int cstride,
                        int K, int act) {
  int lane = threadIdx.x & 31;
  int half = lane >> 4;
  int l16  = lane & 15;
  int tn0 = blockIdx.x * 64;          // first output column of this strip
  int tm  = blockIdx.y;
  v8f acc0 = {}, acc1 = {}, acc2 = {}, acc3 = {};

  const float* arow  = A + (size_t)(tm * 16 + l16) * lda + 2 * half;
  const float* brow0 = B + (size_t)(2 * half) * ldb + tn0 + l16;
  const float* brow1 = brow0 + ldb;
  for (int k = 0; k < K; k += 4) {
    v2f a = *(const v2f*)arow;                     // 8B aligned: lda even
    if (k + 32 < K)                                // stream prefetch (W1 path)
      __builtin_prefetch(brow0 + (size_t)32 * ldb, 0, 1);
    v2f b0, b1, b2, b3;
    b0.x = brow0[0];  b0.y = brow1[0];
    b1.x = brow0[16]; b1.y = brow1[16];
    b2.x = brow0[32]; b2.y = brow1[32];
    b3.x = brow0[48]; b3.y = brow1[48];
    acc0 = __builtin_amdgcn_wmma_f32_16x16x4_f32(false, a, false, b0, (short)0, acc0, false, false);
    acc1 = __builtin_amdgcn_wmma_f32_16x16x4_f32(false, a, false, b1, (short)0, acc1, false, false);
    acc2 = __builtin_amdgcn_wmma_f32_16x16x4_f32(false, a, false, b2, (short)0, acc2, false, false);
    acc3 = __builtin_amdgcn_wmma_f32_16x16x4_f32(false, a, false, b3, (short)0, acc3, false, false);
    arow  += 4;
    brow0 += (size_t)4 * ldb;
    brow1 += (size_t)4 * ldb;
  }
  if (A2) {                                        // fused skip connection
    const float* a2row = A2 + (size_t)(tm * 16 + l16) * lda2 + (size_t)(2 * half) * as2;
    const float* c0 = B2 + (size_t)(2 * half) * ldb + tn0 + l16;
    const float* c1 = c0 + ldb;
    for (int k = 0; k < K; k += 4) {
      v2f a, b0, b1, b2, b3;
      a.x = a2row[0];
      a.y = a2row[as2];
      b0.x = c0[0];  b0.y = c1[0];
      b1.x = c0[16]; b1.y = c1[16];
      b2.x = c0[32]; b2.y = c1[32];
      b3.x = c0[48]; b3.y = c1[48];
      acc0 = __builtin_amdgcn_wmma_f32_16x16x4_f32(false, a, false, b0, (short)0, acc0, false, false);
      acc1 = __builtin_amdgcn_wmma_f32_16x16x4_f32(false, a, false, b1, (short)0, acc1, false, false);
      acc2 = __builtin_amdgcn_wmma_f32_16x16x4_f32(false, a, false, b2, (short)0, acc2, false, false);
      acc3 = __builtin_amdgcn_wmma_f32_16x16x4_f32(false, a, false, b3, (short)0, acc3, false, false);
      a2row += (size_t)4 * as2;
      c0 += (size_t)4 * ldb;
      c1 += (size_t)4 * ldb;
    }
  }
  v8f accs[4] = {acc0, acc1, acc2, acc3};
  for (int tt = 0; tt < 4; tt++) {
    int col = tn0 + tt * 16 + l16;
    float bv = bias ? bias[col] : 0.0f;
    for (int i = 0; i < 8; i++) {
      int row = tm * 16 + i + 8 * half;
      float v = accs[tt][i] + bv;
      if (act == 1) v = tanhf(v);
      C[(size_t)row * ldc + coff + (size_t)col * cstride] = v;
    }
  }
}

// ------------------------------ deterministic CSR by receiver --------------
__global__ void k_hist(const int* __restrict__ recv, int* __restrict__ hist) {
  int ch = blockIdx.x;
  for (int i = threadIdx.x; i < CHSZ; i += blockDim.x) {
    int e = ch * CHSZ + i;
    atomicAdd(&hist[(size_t)ch * NN + recv[e]], 1);
  }
}
__global__ void k_chunkscan(int* __restrict__ hist, int* __restrict__ counts) {
  int n = blockIdx.x * blockDim.x + threadIdx.x;
  if (n >= NN) return;
  int run = 0;
  for (int ch = 0; ch < NCH; ch++) {
    int t = hist[(size_t)ch * NN + n];
    hist[(size_t)ch * NN + n] = run;
    run += t;
  }
  counts[n] = run;
}
__global__ void k_basescan(const int* __restrict__ counts, int* __restrict__ base) {
  __shared__ int part[256];
  int t = threadIdx.x;                 // 256 threads, 25 nodes each
  int s = 0;
  for (int i = 0; i < 25; i++) s += counts[t * 25 + i];
  part[t] = s;
  __syncthreads();
  if (t == 0) {
    int acc = 0;
    for (int i = 0; i < 256; i++) { int tmp = part[i]; part[i] = acc; acc += tmp; }
  }
  __syncthreads();
  int acc = part[t];
  for (int i = 0; i < 25; i++) { base[t * 25 + i] = acc; acc += counts[t * 25 + i]; }
}
__global__ void k_place(const int* __restrict__ recv, const int* __restrict__ hist,
                        const int* __restrict__ base, int* __restrict__ order) {
  __shared__ int lcnt[NN];             // 25.6 KB (<< 320 KB WGP LDS)
  int ch = blockIdx.x;
  for (int i = threadIdx.x; i < NN; i += blockDim.x) lcnt[i] = 0;
  __syncthreads();
  if (threadIdx.x == 0) {
    for (int i = 0; i < CHSZ; i++) {   // sequential => deterministic order
      int e = ch * CHSZ + i;
      int r = recv[e];
      int pos = base[r] + hist[(size_t)ch * NN + r] + lcnt[r]++;
      order[pos] = e;
    }
  }
}

// ------------------------------ edge radial MLP R --------------------------
__global__ void k_redge(const float* __restrict__ ea,
                        const float* __restrict__ Wr1, const float* __restrict__ br1,
                        const float* __restrict__ Wr2, const float* __restrict__ br2,
                        float* __restrict__ R) {
  int e = blockIdx.x;
  int t = threadIdx.x;                 // 64
  __shared__ float ts[32];
  if (t < 32) {
    float u = ea[e] * Wr1[t] + br1[t];
    ts[t] = u / (1.0f + __expf(-u));   // silu
  }
  __syncthreads();
  float s = br2[t];
  for (int j = 0; j < 32; j++) s += ts[j] * Wr2[j * CC + t];
  R[(size_t)e * CC + t] = s;
}

// ------------------------------ Yt = einsum('eb,abd->ead') -----------------
__global__ void k_yt(const float* __restrict__ Y, const float* __restrict__ Wlin,
                     int df, float* __restrict__ Yt) {
  int e = blockIdx.x;
  int t = threadIdx.x;                 // 64
  __shared__ float ys[16];
  if (t < 16) ys[t] = Y[(size_t)e * 16 + t];
  __syncthreads();
  if (t < df * 16) {
    int a = t >> 4, d = t & 15;
    float s = 0.0f;
    for (int b = 0; b < 16; b++) s += ys[b] * Wlin[a * 256 + b * 16 + d];
    Yt[(size_t)e * (df * 16) + t] = s;
  }
}

// ------------------------------ message + ordered segment-sum --------------
// Block per node; 256 threads: thread -> (c = t&63, d-group = t>>6 -> 4 d's).
__global__ void k_msg(const float* __restrict__ R, const float* __restrict__ Yt,
                      const float* __restrict__ hprev, int F, int df,
                      const int* __restrict__ send, const int* __restrict__ order,
                      const int* __restrict__ base, const int* __restrict__ counts,
                      float* __restrict__ A2) {
  int n = blockIdx.x;
  int t = threadIdx.x;
  int c = t & 63;
  int d0 = (t >> 6) * 4;
  float a0 = 0, a1 = 0, a2 = 0, a3 = 0;
  int deg = counts[n], b = base[n];
  for (int i = 0; i < deg; i++) {
    int e = order[b + i];
    int s = send[e];
    float r = R[(size_t)e * CC + c];
    const float* yt = Yt + (size_t)e * (df * 16);
    float fa = hprev[(size_t)s * F + c];                 // a = 0
    float S0 = fa * yt[d0], S1 = fa * yt[d0 + 1];
    float S2 = fa * yt[d0 + 2], S3 = fa * yt[d0 + 3];
    for (int a = 1; a < df; a++) {
      float fv = hprev[(size_t)s * F + 64 + c * 3 + (a - 1)];
      const float* ya = yt + a * 16;
      S0 += fv * ya[d0]; S1 += fv * ya[d0 + 1];
      S2 += fv * ya[d0 + 2]; S3 += fv * ya[d0 + 3];
    }
    a0 += r * S0; a1 += r * S1; a2 += r * S2; a3 += r * S3;
  }
  const float inv = 0.1f;              // 1/AVG_NEIGH
  float* o = A2 + ((size_t)n * CC + c) * 16 + d0;
  o[0] = a0 * inv; o[1] = a1 * inv; o[2] = a2 * inv; o[3] = a3 * inv;
}

// ------------------------------ per-node nonlinearity ----------------------
__global__ void k_node(const float* __restrict__ A2, const float* __restrict__ wz,
                       int K, float* __restrict__ scal, float* __restrict__ vec) {
  int t = blockIdx.x * blockDim.x + threadIdx.x;
  if (t >= NN * CC) return;
  int n = t >> 6, c = t & 63;
  const float* a = A2 + (size_t)t * 16;
  float s1 = a[0];
  float v0 = a[1], v1 = a[2], v2 = a[3];
  float s2 = 0, s3 = 0;
  for (int d = 0; d < 16; d++) { float x = a[d]; s2 += x * x; s3 += x * x * x; }
  const float* w = wz + (size_t)n * CC * K + c * K;
  float sc = w[0] * s1 + w[1] * s1 * s1 + w[2] * s2 + w[3] * s1 * s1 * s1
           + w[4] * s1 * s2 + w[5] * s3;
  scal[t] = sc;
  if (K == 9) {
    float coef = w[6] + s1 * w[7] + s2 * w[8];
    vec[0 * (size_t)NN * CC + t] = v0 * coef;
    vec[1 * (size_t)NN * CC + t] = v1 * coef;
    vec[2 * (size_t)NN * CC + t] = v2 * coef;
  }
}

// ------------------------------ readout ------------------------------------
__global__ void k_gmean(const float* __restrict__ h1, const float* __restrict__ h2,
                        const float* __restrict__ h3, const float* __restrict__ h4,
                        float* __restrict__ hcat) {
  int t = blockIdx.x * blockDim.x + threadIdx.x;
  if (t >= NGR * DH) return;
  int g = t / DH, f = t % DH;
  float s = 0;
  for (int i = 0; i < AA; i++) {
    int n = g * AA + i;
    float v;
    if      (f < 64)  v = h1[(size_t)n * 64  + f];
    else if (f < 320) v = h2[(size_t)n * 256 + (f - 64)];
    else if (f < 576) v = h3[(size_t)n * 256 + (f - 320)];
    else              v = h4[(size_t)n * 64  + (f - 576)];
    s += v;
  }
  hcat[t] = s * (1.0f / AA);
}

__global__ void k_bnh(const float* __restrict__ hcat, const float* __restrict__ g,
                      const float* __restrict__ b, float* __restrict__ z) {
  int f = blockIdx.x * blockDim.x + threadIdx.x;
  if (f >= DH) return;
  float m = 0;
  for (int r = 0; r < NGR; r++) m += hcat[r * DH + f];
  m *= (1.0f / NGR);
  float v = 0;
  for (int r = 0; r < NGR; r++) { float d = hcat[r * DH + f] - m; v += d * d; }
  v *= (1.0f / NGR);
  float is = rsqrtf(v + 1e-5f);
  for (int r = 0; r < NGR; r++)
    z[(size_t)r * DIN + D1 + f] = (hcat[r * DH + f] - m) * is * g[f] + b[f];
}

__global__ void k_bnxt(const float* __restrict__ x, const float* __restrict__ g,
                       const float* __restrict__ b, float* __restrict__ z) {
  int p = blockIdx.x * blockDim.x + threadIdx.x;
  if (p >= D1) return;
  // invert p -> (i,j) with j>i from upper-triangle (k=1) row-major order
  float disc = 799.0f * 799.0f - 8.0f * (float)p;
  int i = (int)((799.0f - sqrtf(fmaxf(disc, 0.0f))) * 0.5f);
  if (i < 0) i = 0; if (i > 398) i = 398;
  long long off = (long long)i * 399 - (long long)i * (i - 1) / 2;
  while (off > p) { i--; off = (long long)i * 399 - (long long)i * (i - 1) / 2; }
  while (off + (399 - i) <= p) { off += (399 - i); i++; }
  int j = i + 1 + (int)(p - off);
  float vals[NGR];
  float m = 0;
  for (int r = 0; r < NGR; r++) {
    vals[r] = x[((size_t)(r * AA + i)) * AA + j];
    m += vals[r];
  }
  m *= (1.0f / NGR);
  float v = 0;
  for (int r = 0; r < NGR; r++) { float d = vals[r] - m; v += d * d; }
  v *= (1.0f / NGR);
  float is = rsqrtf(v + 1e-5f);
  for (int r = 0; r < NGR; r++)
    z[(size_t)r * DIN + p] = (vals[r] - m) * is * g[p] + b[p];
}

__global__ void k_bnrelu(float* __restrict__ Z, const float* __restrict__ g,
                         const float* __restrict__ b, int ncol) {
  int c = blockIdx.x * blockDim.x + threadIdx.x;
  if (c >= ncol) return;
  float m = 0;
  for (int r = 0; r < NGR; r++) m += Z[r * ncol + c];
  m *= (1.0f / NGR);
  float v = 0;
  for (int r = 0; r < NGR; r++) { float d = Z[r * ncol + c] - m; v += d * d; }
  v *= (1.0f / NGR);
  float is = rsqrtf(v + 1e-5f);
  for (int r = 0; r < NGR; r++) {
    float val = (Z[r * ncol + c] - m) * is * g[c] + b[c];
    Z[r * ncol + c] = fmaxf(val, 0.0f);
  }
}

__global__ void k_head(const float* __restrict__ z3, const float* __restrict__ W4,
                       const float* __restrict__ b4, float* __restrict__ out) {
  int r = threadIdx.x;
  if (r >= NGR) return;
  float o[7];
  for (int k = 0; k < 7; k++) o[k] = b4[k];
  for (int j = 0; j < 256; j++) {
    float zj = z3[r * 256 + j];
    for (int k = 0; k < 7; k++) o[k] += zj * W4[j * 7 + k];
  }
  float m = o[0];
  for (int k = 1; k < 7; k++) m = fmaxf(m, o[k]);
  float s = 0;
  for (int k = 0; k < 7; k++) s += expf(o[k] - m);
  float lse = m + logf(s);
  for (int k = 0; k < 7; k++) out[r * 7 + k] = o[k] - lse;
}

// ---------------------------------------------------------------------------
static inline void gemm(hipStream_t st, const float* A, int lda,
                        const float* B, int ldb,
                        const float* A2, int lda2, int as2, const float* B2,
                        const float* bias, float* C, int ldc, int coff, int cs,
                        int M, int N, int K, int act) {
  dim3 g(N / 64, M / 16);              // all N dims here are multiples of 64
  k_gemm4<<<g, 32, 0, st>>>(A, lda, B, ldb, A2, lda2, as2, B2,
                            bias, C, ldc, coff, cs, K, act);
}

extern "C" void kernel_launch(void* const* d_in, const int* in_sizes, int n_in,
                              void* d_out, int out_size, void* d_ws, size_t ws_size,
                              hipStream_t stream) {
  // ---- input mapping: setup_inputs() dict insertion order, params nested in
  //      insertion order (layers list in order; mlp dict in insertion order).
  int i = 0;
  const float* x    = (const float*)d_in[i++];   // (6400,400)
  const float* ev   = (const float*)d_in[i++];   // (3,E)
  const float* ea   = (const float*)d_in[i++];   // (E,1)
  const int*   eidx = (const int*)  d_in[i++];   // (2,E): [0]=senders [1]=receivers
  const float* Wemb = (const float*)d_in[i++];   // (400,64)
  struct LP { const float *Wr1,*br1,*Wr2,*br2,*Wlin,*Wattr,*Wm0,*Wsc0,*Wm1,*Wsc1;
              int df, K; };
  LP L[3];
  for (int l = 0; l < 3; l++) {
    L[l].Wr1  = (const float*)d_in[i++];
    L[l].br1  = (const float*)d_in[i++];
    L[l].Wr2  = (const float*)d_in[i++];
    L[l].br2  = (const float*)d_in[i++];
    L[l].Wlin = (const float*)d_in[i++];
    L[l].Wattr= (const float*)d_in[i++];
    L[l].Wm0  = (const float*)d_in[i++];
    L[l].Wsc0 = (const float*)d_in[i++];
    L[l].Wm1  = (l < 2) ? (const float*)d_in[i++] : nullptr;
    L[l].Wsc1 = (l == 1) ? (const float*)d_in[i++] : nullptr;
    L[l].df = (l == 0) ? 1 : 4;
    L[l].K  = (l == 2) ? 6 : 9;
  }
  const float* bn_g  = (const float*)d_in[i++];
  const float* bn_b  = (const float*)d_in[i++];
  const float* bnh_g = (const float*)d_in[i++];
  const float* bnh_b = (const float*)d_in[i++];
  const float* W1 = (const float*)d_in[i++]; const float* b1 = (const float*)d_in[i++];
  const float* g1 = (const float*)d_in[i++]; const float* be1= (const float*)d_in[i++];
  const float* W2 = (const float*)d_in[i++]; const float* b2 = (const float*)d_in[i++];
  const float* g2 = (const float*)d_in[i++]; const float* be2= (const float*)d_in[i++];
  const float* W3 = (const float*)d_in[i++]; const float* b3 = (const float*)d_in[i++];
  const float* g3 = (const float*)d_in[i++]; const float* be3= (const float*)d_in[i++];
  const float* W4 = (const float*)d_in[i++]; const float* b4 = (const float*)d_in[i++];
  (void)in_sizes; (void)n_in; (void)out_size; (void)ws_size;

  // ---- workspace carve (≈108 MB total)
  char* p = (char*)d_ws;
  auto carve = [&](size_t bytes) -> void* {
    void* r = (void*)p; p += (bytes + 255) & ~(size_t)255; return r;
  };
  float* Y     = (float*)carve((size_t)EE * 16 * 4);
  float* Yt    = (float*)carve((size_t)EE * 64 * 4);
  float* R     = (float*)carve((size_t)EE * 64 * 4);
  float* h1    = (float*)carve((size_t)NN * 64 * 4);
  float* h2    = (float*)carve((size_t)NN * 256 * 4);
  float* h3    = (float*)carve((size_t)NN * 256 * 4);
  float* h4    = (float*)carve((size_t)NN * 64 * 4);
  float* A2    = (float*)carve((size_t)NN * 1024 * 4);
  float* wz    = (float*)carve((size_t)NN * 576 * 4);
  float* scal  = (float*)carve((size_t)NN * 64 * 4);
  float* vec   = (float*)carve((size_t)3 * NN * 64 * 4);
  int*   hist  = (int*)  carve((size_t)NCH * NN * 4);
  int*   counts= (int*)  carve((size_t)NN * 4);
  int*   base  = (int*)  carve((size_t)NN * 4);
  int*   order = (int*)  carve((size_t)EE * 4);
  float* hcat  = (float*)carve((size_t)NGR * DH * 4);
  float* z     = (float*)carve((size_t)NGR * DIN * 4);
  float* z1    = (float*)carve((size_t)NGR * 512 * 4);
  float* z2    = (float*)carve((size_t)NGR * 256 * 4);
  float* z3    = (float*)carve((size_t)NGR * 256 * 4);

  const int* send = eidx;
  const int* recv = eidx + EE;

  // ---- deterministic CSR of edges by receiver
  hipMemsetAsync(hist, 0, (size_t)NCH * NN * 4, stream);
  k_hist<<<NCH, 256, 0, stream>>>(recv, hist);
  k_chunkscan<<<(NN + 255) / 256, 256, 0, stream>>>(hist, counts);
  k_basescan<<<1, 256, 0, stream>>>(counts, base);
  k_place<<<NCH, 256, 0, stream>>>(recv, hist, base, order);

  // ---- spherical harmonics
  k_sph<<<EE / 256, 256, 0, stream>>>(ev, Y);

  // ---- node embedding: h1 = tanh(x @ W_embed)
  gemm(stream, x, AA, Wemb, CC, nullptr, 0, 0, nullptr, nullptr,
       h1, CC, 0, 1, NN, CC, AA, 1);

  // ---- three MACE layers
  const float* hprev = h1;
  int Fprev = 64;
  float* houts[3] = {h2, h3, h4};
  int Fouts[3]    = {256, 256, 64};
  for (int l = 0; l < 3; l++) {
    const LP& P = L[l];
    int CK = CC * P.K;
    // wz = x @ Wattr     (6400 x 400 x CK)  -- the big per-layer GEMM
    gemm(stream, x, AA, P.Wattr, CK, nullptr, 0, 0, nullptr, nullptr,
         wz, CK, 0, 1, NN, CK, AA, 0);
    // R = (silu(ea@Wr1+br1))@Wr2 + br2
    k_redge<<<EE, 64, 0, stream>>>(ea, P.Wr1, P.br1, P.Wr2, P.br2, R);
    // Yt = einsum('eb,abd->ead')
    k_yt<<<EE, 64, 0, stream>>>(Y, P.Wlin, P.df, Yt);
    // messages + ordered segment-sum -> A2 (N,64,16)
    k_msg<<<NN, 256, 0, stream>>>(R, Yt, hprev, Fprev, P.df, send, order,
                                  base, counts, A2);
    // nonlinear body -> scal (and vec if K==9)
    k_node<<<(NN * CC + 255) / 256, 256, 0, stream>>>(A2, wz, P.K, scal, vec);
    // out0 = tanh(scal@Wm0 + f0@Wsc0)
    float* hn = houts[l];
    int Fn = Fouts[l];
    gemm(stream, scal, CC, P.Wm0, CC, hprev, Fprev, 1, P.Wsc0, nullptr,
         hn, Fn, 0, 1, NN, CC, CC, 1);
    // out1_v = tanh(vec_v@Wm1 [+ fvec_v@Wsc1]) interleaved at cols 64+3c+v
    if (P.K == 9) {
      for (int v = 0; v < 3; v++) {
        const float* Asc = P.Wsc1 ? (hprev + 64 + v) : nullptr;
        gemm(stream, vec + (size_t)v * NN * CC, CC, P.Wm1,
             CC, Asc, Fprev, 3, P.Wsc1, nullptr,
             hn, Fn, 64 + v, 3, NN, CC, CC, 1);
      }
    }
    hprev = hn;
    Fprev = Fn;
  }

  // ---- readout: graph means + batchnorms into z (16 x 80440)
  k_gmean<<<(NGR * DH + 255) / 256, 256, 0, stream>>>(h1, h2, h3, h4, hcat);
  k_bnh<<<(DH + 255) / 256, 256, 0, stream>>>(hcat, bnh_g, bnh_b, z);
  k_bnxt<<<(D1 + 255) / 256, 256, 0, stream>>>(x, bn_g, bn_b, z);

  // ---- MLP head (W1 GEMM is the bandwidth hot spot: 165 MB of weights)
  gemm(stream, z, DIN, W1, 512, nullptr, 0, 0, nullptr, b1,
       z1, 512, 0, 1, NGR, 512, DIN, 0);
  k_bnrelu<<<2, 256, 0, stream>>>(z1, g1, be1, 512);
  gemm(stream, z1, 512, W2, 256, nullptr, 0, 0, nullptr, b2,
       z2, 256, 0, 1, NGR, 256, 512, 0);
  k_bnrelu<<<1, 256, 0, stream>>>(z2, g2, be2, 256);
  gemm(stream, z2, 256, W3, 256, nullptr, 0, 0, nullptr, b3,
       z3, 256, 0, 1, NGR, 256, 256, 0);
  k_bnrelu<<<1, 256, 0, stream>>>(z3, g3, be3, 256);
  k_head<<<1, 32, 0, stream>>>(z3, W4, b4, (float*)d_out);
}